// hybrid_position_embedding_17549236371827
// MI455X (gfx1250) — compile-verified
//
#include <hip/hip_runtime.h>
#include <hip/hip_bf16.h>

// Reference collapses algebraically:
//   out = softmax(hybrid[:, None, :], axis=1)  with axis-1 length == 1
// => out == 1.0f everywhere (hybrid is finite after the int32 round-trip).
// Optimal kernel = stream 1 MiB of 1.0f to d_out with 128-bit NT stores.

typedef float v4f __attribute__((ext_vector_type(4)));

__global__ __launch_bounds__(256)
void hpe_fill_ones_b128(v4f* __restrict__ out, int n4) {
    int i = blockIdx.x * blockDim.x + threadIdx.x;
    if (i < n4) {
        v4f v = {1.0f, 1.0f, 1.0f, 1.0f};
        // Non-temporal: pure streaming write, skip cache retention.
        __builtin_nontemporal_store(v, &out[i]);
    }
}

__global__ __launch_bounds__(32)
void hpe_fill_ones_tail(float* __restrict__ out, int n) {
    int i = threadIdx.x;
    if (i < n) out[i] = 1.0f;
}

extern "C" void kernel_launch(void* const* d_in, const int* in_sizes, int n_in,
                              void* d_out, int out_size, void* d_ws, size_t ws_size,
                              hipStream_t stream) {
    (void)d_in; (void)in_sizes; (void)n_in; (void)d_ws; (void)ws_size;

    float* out = (float*)d_out;              // reference output dtype: float32
    int n4  = out_size >> 2;                 // 262144 / 4 = 65536 float4 stores
    int rem = out_size & 3;                  // 0 for this shape, kept for safety

    if (n4 > 0) {
        const int threads = 256;             // 8 wave32s per block
        const int blocks  = (n4 + threads - 1) / threads;
        hpe_fill_ones_b128<<<blocks, threads, 0, stream>>>((v4f*)out, n4);
    }
    if (rem > 0) {
        hpe_fill_ones_tail<<<1, 32, 0, stream>>>(out + (size_t)n4 * 4, rem);
    }
}